// ClusterEncoder_54511724921261
// MI455X (gfx1250) — compile-verified
//
#include <hip/hip_runtime.h>
#include <hip/hip_bf16.h>

// Problem constants from the reference
#define Bb 64
#define Pp 10000
#define Ee 128
#define Cc 100
#define SPLIT 8                          // P-chunks per batch (privatized LDS accumulators)
#define CHUNK ((Pp + SPLIT - 1) / SPLIT) // 1250 nodes per block

typedef __attribute__((ext_vector_type(2))) float v2f;
typedef __attribute__((ext_vector_type(8))) float v8f;

// ---------------------------------------------------------------------------
// Pass 1: per-batch segment-sum (scatter-add) with LDS privatization.
// One wave streams one node row (32 lanes x float4 = 128 floats = 512B),
// ds_add_f32 into the block-private [C][E] accumulator, then one flush of
// LDS -> global via hardware f32 atomics.
// ---------------------------------------------------------------------------
__global__ __launch_bounds__(256) void pass1_scatter(
    const float* __restrict__ enc, const int* __restrict__ ids,
    float* __restrict__ ws_sum, float* __restrict__ ws_cnt)
{
    __shared__ float s_sum[Cc * Ee];   // 51200 B
    __shared__ float s_cnt[Cc];

    const int tid   = threadIdx.x;
    const int b     = blockIdx.x / SPLIT;
    const int chunk = blockIdx.x % SPLIT;

    for (int i = tid; i < Cc * Ee; i += 256) s_sum[i] = 0.0f;
    for (int i = tid; i < Cc;     i += 256) s_cnt[i] = 0.0f;
    __syncthreads();

    const int lane = tid & 31;
    const int wave = tid >> 5;          // 8 waves per block

    int n_end = (chunk + 1) * CHUNK;
    if (n_end > Pp) n_end = Pp;

    for (int n = chunk * CHUNK + wave; n < n_end; n += 8) {
        const size_t row = (size_t)b * Pp + n;
        const int    c   = ids[row];
        const float4 v   = ((const float4*)(enc + row * Ee))[lane];
        float* dst = &s_sum[c * Ee + lane * 4];
        atomicAdd(dst + 0, v.x);        // ds_add_f32
        atomicAdd(dst + 1, v.y);
        atomicAdd(dst + 2, v.z);
        atomicAdd(dst + 3, v.w);
        if (lane == 0) atomicAdd(&s_cnt[c], 1.0f);
    }
    __syncthreads();

    float* gs = ws_sum + (size_t)b * Cc * Ee;
    for (int i = tid; i < Cc * Ee; i += 256)
        unsafeAtomicAdd(&gs[i], s_sum[i]);          // global_atomic_add_f32
    for (int i = tid; i < Cc; i += 256)
        unsafeAtomicAdd(&ws_cnt[b * Cc + i], s_cnt[i]);
}

// ---------------------------------------------------------------------------
// Pass 2: cluster_embs = (sums / max(count,1)) @ W^T + bias using
// V_WMMA_F32_16X16X4_F32 (f32 -> exact precision vs f32 reference).
// Block = 16 flattened (b*C+c) rows; 8 waves = 8 N-tiles of 16 columns.
// A fragment (16x4 f32, 2 VGPR): lanes 0-15 -> K pair {0,1}, lanes 16-31 -> {2,3}.
// B fragment (4x16 f32, 2 VGPR): same K split; N = lane%16.
// D (16x16 f32, 8 VGPR): VGPR r holds M=r (lanes 0-15) / M=r+8 (lanes 16-31).
// ---------------------------------------------------------------------------
__global__ __launch_bounds__(256) void pass2_mean_gemm(
    const float* __restrict__ ws_sum, const float* __restrict__ ws_cnt,
    const float* __restrict__ W, const float* __restrict__ bias,
    float* __restrict__ out_ce)
{
    __shared__ float s_mean[16 * Ee];  // 8 KB, A-tile shared by all 8 waves
    __shared__ float s_inv[16];

    const int tid  = threadIdx.x;
    const int row0 = blockIdx.x * 16;  // rows flattened over B*C = 6400

    if (tid < 16) s_inv[tid] = 1.0f / fmaxf(ws_cnt[row0 + tid], 1.0f);
    __syncthreads();
    for (int i = tid; i < 16 * Ee; i += 256)
        s_mean[i] = ws_sum[(size_t)row0 * Ee + i] * s_inv[i >> 7];
    __syncthreads();

    const int lane = tid & 31;
    const int wave = tid >> 5;
    const int n0   = wave * 16;
    const int mrow = lane & 15;
    const int klo  = (lane < 16) ? 0 : 2;

    v8f acc = {};
#pragma unroll
    for (int kk = 0; kk < Ee / 4; ++kk) {
        const int k = kk * 4 + klo;
        v2f a = *(const v2f*)&s_mean[mrow * Ee + k];
        v2f bf = *(const v2f*)&W[(size_t)(n0 + mrow) * Ee + k];
        acc = __builtin_amdgcn_wmma_f32_16x16x4_f32(
            /*neg_a=*/false, a, /*neg_b=*/false, bf,
            /*c_mod=*/(short)0, acc, /*reuse_a=*/false, /*reuse_b=*/false);
    }

    const int   j     = n0 + mrow;
    const float pb    = bias[j];
    const int   mbase = (lane < 16) ? 0 : 8;
#pragma unroll
    for (int r = 0; r < 8; ++r) {
        const int m = mbase + r;
        out_ce[(size_t)(row0 + m) * Ee + j] = acc[r] + pb;
    }
}

// ---------------------------------------------------------------------------
// Pass 3: g_node[b,p,:] = cluster_embs[b, ids[b,p], :]
// One wave per node row: broadcast-gather 512B (L2-resident, 3.3MB working
// set), streaming float4 store.
// ---------------------------------------------------------------------------
__global__ __launch_bounds__(256) void pass3_gather(
    const float* __restrict__ ce, const int* __restrict__ ids,
    float* __restrict__ g_node)
{
    const int    lane = threadIdx.x & 31;
    const int    wave = threadIdx.x >> 5;
    const size_t row  = (size_t)blockIdx.x * 8 + wave;   // b*P + p
    const int    b    = (int)(row / Pp);
    const int    c    = ids[row];
    const float4 v = ((const float4*)(ce + (size_t)(b * Cc + c) * Ee))[lane];
    ((float4*)(g_node + row * Ee))[lane] = v;
}

extern "C" void kernel_launch(void* const* d_in, const int* in_sizes, int n_in,
                              void* d_out, int out_size, void* d_ws, size_t ws_size,
                              hipStream_t stream) {
    const float* enc  = (const float*)d_in[0];   // [B,P,E] f32
    const int*   ids  = (const int*)d_in[1];     // [B,P] int
    const float* W    = (const float*)d_in[3];   // [E,E] f32
    const float* bias = (const float*)d_in[4];   // [E]   f32

    float* out    = (float*)d_out;
    float* out_ce = out;                               // cluster_embs [B,C,E]
    float* out_gn = out + (size_t)Bb * Cc * Ee;        // g_node       [B,P,E]

    float* ws_sum = (float*)d_ws;                      // [B,C,E]
    float* ws_cnt = ws_sum + (size_t)Bb * Cc * Ee;     // [B,C]

    hipMemsetAsync(d_ws, 0,
                   ((size_t)Bb * Cc * Ee + (size_t)Bb * Cc) * sizeof(float),
                   stream);

    pass1_scatter<<<Bb * SPLIT, 256, 0, stream>>>(enc, ids, ws_sum, ws_cnt);
    pass2_mean_gemm<<<(Bb * Cc) / 16, 256, 0, stream>>>(ws_sum, ws_cnt, W, bias, out_ce);
    pass3_gather<<<(Bb * Pp) / 8, 256, 0, stream>>>(out_ce, ids, out_gn);
}